// InfoNCELoss_55173149884647
// MI455X (gfx1250) — compile-verified
//
#include <hip/hip_runtime.h>

// InfoNCE loss, B=16 N=4096 D=1024, A ~= 65536 anchors.
// Memory-bound gather (~768 MB raw, ~33us at 23.3 TB/s HBM).
// WMMA f32 16x16x4 with an all-ones B matrix is used as a cross-lane
// row-sum engine: C += (a .* p) x ones  accumulates the 16 per-anchor dot
// products into the C/D diagonal-free layout (every N column identical),
// exactly in f32, with no dependence on the K-slot mapping of the A tile.

typedef __attribute__((ext_vector_type(2))) float v2f;
typedef __attribute__((ext_vector_type(8))) float v8f;

#define D_DIM   1024
#define INV_T   10.0f   // 1 / TEMPERATURE

__global__ __launch_bounds__(256) void infonce_wmma_kernel(
    const float* __restrict__ emb,
    const int*   __restrict__ aidx,
    const int*   __restrict__ pidx,
    const int*   __restrict__ nidx,
    int A,
    float* __restrict__ partials)
{
    __shared__ float red[256];
    const int tid  = threadIdx.x;
    const int lane = tid & 31;
    const int gwave = blockIdx.x * (blockDim.x >> 5) + (tid >> 5);
    const int aBase = gwave * 16;

    float partial = 0.0f;

    if (aBase < A) {                       // wave-uniform guard: EXEC stays all-1s
        const int row  = lane & 15;        // anchor row this lane serves
        const int half = lane >> 4;        // which K-half of the row
        const int m    = aBase + row;
        const int mi   = (m < A) ? m : (A - 1);   // clamp tail (contribution masked later)

        const float* __restrict__ arow = emb + (long long)aidx[mi] * D_DIM;
        const float* __restrict__ prow = emb + (long long)pidx[mi] * D_DIM;
        const float* __restrict__ nrow = emb + (long long)nidx[mi] * D_DIM;

        v8f cpos = {};
        v8f cneg = {};
        const v2f ones = {1.0f, 1.0f};     // B matrix: all ones -> layout-immune row sum

        // Each macro-iter: this lane consumes 4 consecutive K elements; the two
        // halves of the row interleave 16B chunks (32B contiguous per row/iter).
        #pragma unroll 2
        for (int t = 0; t < D_DIM / 8; ++t) {
            const int k = t * 8 + half * 4;
            const float4 a4 = *(const float4*)(arow + k);   // global_load_b128
            const float4 p4 = *(const float4*)(prow + k);
            const float4 n4 = *(const float4*)(nrow + k);

            v2f ap0 = { a4.x * p4.x, a4.y * p4.y };
            v2f an0 = { a4.x * n4.x, a4.y * n4.y };
            v2f ap1 = { a4.z * p4.z, a4.w * p4.w };
            v2f an1 = { a4.z * n4.z, a4.w * n4.w };

            // interleave the two accumulator chains to pipeline the XDL
            cpos = __builtin_amdgcn_wmma_f32_16x16x4_f32(false, ap0, false, ones,
                                                         (short)0, cpos, false, false);
            cneg = __builtin_amdgcn_wmma_f32_16x16x4_f32(false, an0, false, ones,
                                                         (short)0, cneg, false, false);
            cpos = __builtin_amdgcn_wmma_f32_16x16x4_f32(false, ap1, false, ones,
                                                         (short)0, cpos, false, false);
            cneg = __builtin_amdgcn_wmma_f32_16x16x4_f32(false, an1, false, ones,
                                                         (short)0, cneg, false, false);
        }

        // C/D layout: lanes 0-15 VGPR j -> M=j ; lanes 16-31 VGPR j -> M=8+j.
        // Lane 0 extracts rows 0..7, lane 16 extracts rows 8..15 (N columns identical).
        if ((lane & 15) == 0) {
            const int mb = aBase + half * 8;
            #pragma unroll
            for (int j = 0; j < 8; ++j) {
                if (mb + j < A) {
                    const float ps = cpos[j] * INV_T;
                    const float ns = cneg[j] * INV_T;
                    const float mx = fmaxf(ps, ns);
                    const float lse = mx + __logf(__expf(ps - mx) + __expf(ns - mx));
                    partial += lse - ps;
                }
            }
        }
    }

    // deterministic block reduction -> one partial per block
    red[tid] = partial;
    __syncthreads();
    #pragma unroll
    for (int s = 128; s > 0; s >>= 1) {
        if (tid < s) red[tid] += red[tid + s];
        __syncthreads();
    }
    if (tid == 0) partials[blockIdx.x] = red[0];
}

__global__ __launch_bounds__(256) void infonce_reduce_kernel(
    const float* __restrict__ partials, int n, float invA, float* __restrict__ out)
{
    __shared__ float red[256];
    float s = 0.0f;
    for (int i = threadIdx.x; i < n; i += 256) s += partials[i];
    red[threadIdx.x] = s;
    __syncthreads();
    #pragma unroll
    for (int t = 128; t > 0; t >>= 1) {
        if (threadIdx.x < t) red[threadIdx.x] += red[threadIdx.x + t];
        __syncthreads();
    }
    if (threadIdx.x == 0) out[0] = red[0] * invA;
}

extern "C" void kernel_launch(void* const* d_in, const int* in_sizes, int n_in,
                              void* d_out, int out_size, void* d_ws, size_t ws_size,
                              hipStream_t stream) {
    const float* emb  = (const float*)d_in[0];
    const int*   aidx = (const int*)d_in[1];
    const int*   pidx = (const int*)d_in[2];
    const int*   nidx = (const int*)d_in[3];
    const int A = in_sizes[1];

    const int wavesPerBlock = 8;                         // 256 threads = 8 wave32
    const int waves  = (A + 15) / 16;                    // 16 anchors per wave
    const int blocks = (waves + wavesPerBlock - 1) / wavesPerBlock;

    float* partials = (float*)d_ws;

    infonce_wmma_kernel<<<blocks, 256, 0, stream>>>(emb, aidx, pidx, nidx, A, partials);
    infonce_reduce_kernel<<<1, 256, 0, stream>>>(partials, blocks, 1.0f / (float)A,
                                                 (float*)d_out);
}